// SelfAttention_8100308320902
// MI455X (gfx1250) — compile-verified
//
#include <hip/hip_runtime.h>
#include <math.h>

typedef float v2f __attribute__((ext_vector_type(2)));
typedef float v8f __attribute__((ext_vector_type(8)));

#define N_KP 24     // keypoints
#define DDIM 128    // feature dim
#define FS   132    // LDS row stride for f / h (bank-conflict padding, 16B-friendly)
#define WS   26     // LDS row stride for w1/w2/p (even -> 8B-aligned float2 frags)
#define SS   27     // LDS row stride for attention scores (odd -> conflict-free rows)

// D = A(16x4 f32) * B(4x16 f32) + C(16x16 f32), full fp32 precision on the matrix pipe
static __device__ __forceinline__ v8f wmma_f32(v2f a, v2f b, v8f c) {
  return __builtin_amdgcn_wmma_f32_16x16x4_f32(false, a, false, b, (short)0, c,
                                               false, false);
}

// Row-major [row][k] fragment: lane L holds (row = base+L%16) x (k = k0 + 2*(L/16) .. +1).
// Serves both the A operand (M x K) and, for S = W*f^T, the B operand (since
// B[k][n] = f[n][k] has the identical per-lane addressing). Rows >= 24 are zero-padding.
static __device__ __forceinline__ v2f fragRK(const float* __restrict__ s,
                                             int row, int k, int stride) {
  v2f r;
  if (row < N_KP) {
    r.x = s[row * stride + k];
    r.y = s[row * stride + k + 1];
  } else {
    r.x = 0.0f;
    r.y = 0.0f;
  }
  return r;
}

__global__ __launch_bounds__(64) void attn24_kernel(
    const float* __restrict__ F, const float* __restrict__ W1,
    const float* __restrict__ B1v, const float* __restrict__ W2,
    const float* __restrict__ B2v, float* __restrict__ O, int nbatch) {
  __shared__ __align__(16) float w1s[N_KP * WS];
  __shared__ __align__(16) float w2s[N_KP * WS];
  __shared__ float b1s[N_KP];
  __shared__ float b2s[N_KP];
  __shared__ __align__(16) float fbuf[2][N_KP * FS];
  __shared__ __align__(16) float hbuf[2][N_KP * FS];

  const int tid = threadIdx.x;

  // Cooperative load of the tiny shared weights (24x24 each) + biases.
  for (int i = tid; i < N_KP * N_KP; i += 64) {
    int m = i / N_KP;
    int k = i - m * N_KP;
    w1s[m * WS + k] = W1[i];
    w2s[m * WS + k] = W2[i];
  }
  if (tid < N_KP) {
    b1s[tid] = B1v[tid];
    b2s[tid] = B2v[tid];
  }
  __syncthreads();

  const int wv   = tid >> 5;   // wave index in block (one batch item per wave)
  const int lane = tid & 31;
  const int half = lane >> 4;  // 0: lanes 0-15, 1: lanes 16-31
  const int lm   = lane & 15;
  const int koff = half << 1;  // K sub-offset per WMMA f32 frag layout

  const int b = blockIdx.x * 2 + wv;
  if (b >= nbatch) return;  // wave-uniform

  float* fl = fbuf[wv];
  float* hl = hbuf[wv];
  const float* gf = F + (size_t)b * (N_KP * DDIM);

  // ---- Stage 0: f -> LDS (coalesced float4, 768 float4 per item) ----
  #pragma unroll
  for (int i = 0; i < 24; ++i) {
    int idx = lane + 32 * i;       // 0..767
    int row = idx >> 5;            // 32 float4 per 128-wide row
    int c4  = idx & 31;
    float4 v = reinterpret_cast<const float4*>(gf)[idx];
    *reinterpret_cast<float4*>(&fl[row * FS + (c4 << 2)]) = v;
  }

  // ---- Stage 1: h = relu(w1 @ f + b1), h -> LDS ----
  {
    v2f a1[2][6];
    #pragma unroll
    for (int mt = 0; mt < 2; ++mt)
      #pragma unroll
      for (int kk = 0; kk < 6; ++kk)
        a1[mt][kk] = fragRK(w1s, mt * 16 + lm, kk * 4 + koff, WS);

    for (int nt = 0; nt < 8; ++nt) {
      int col = nt * 16 + lm;
      #pragma unroll
      for (int mt = 0; mt < 2; ++mt) {
        v8f acc = {};
        #pragma unroll
        for (int kk = 0; kk < 6; ++kk) {
          int kb = kk * 4 + koff;
          v2f bf;
          bf.x = fl[kb * FS + col];
          bf.y = fl[(kb + 1) * FS + col];
          acc = wmma_f32(a1[mt][kk], bf, acc);
        }
        #pragma unroll
        for (int r = 0; r < 8; ++r) {
          int m = mt * 16 + half * 8 + r;
          if (m < N_KP) {
            float v = acc[r] + b1s[m];
            hl[m * FS + col] = v > 0.0f ? v : 0.0f;
          }
        }
      }
    }
  }

  // ---- Stage 2: weights = w2 @ h + b2, overwrites h in-place per column tile ----
  {
    v2f a2[2][6];
    #pragma unroll
    for (int mt = 0; mt < 2; ++mt)
      #pragma unroll
      for (int kk = 0; kk < 6; ++kk)
        a2[mt][kk] = fragRK(w2s, mt * 16 + lm, kk * 4 + koff, WS);

    for (int nt = 0; nt < 8; ++nt) {
      int col = nt * 16 + lm;
      v8f acc0 = {}, acc1 = {};
      #pragma unroll
      for (int kk = 0; kk < 6; ++kk) {
        int kb = kk * 4 + koff;
        v2f bf;
        bf.x = hl[kb * FS + col];
        bf.y = hl[(kb + 1) * FS + col];
        acc0 = wmma_f32(a2[0][kk], bf, acc0);
        acc1 = wmma_f32(a2[1][kk], bf, acc1);
      }
      #pragma unroll
      for (int r = 0; r < 8; ++r) {
        int m0 = half * 8 + r;          // < 16, always valid
        hl[m0 * FS + col] = acc0[r] + b2s[m0];
        int m1 = 16 + half * 8 + r;
        if (m1 < N_KP) hl[m1 * FS + col] = acc1[r] + b2s[m1];
      }
    }
  }

  // ---- Stage 3: S = weights @ f^T  (24x24 in four 16x16 tiles, K=128) ----
  v8f s00 = {}, s01 = {}, s10 = {}, s11 = {};
  for (int kk = 0; kk < 32; ++kk) {
    int kb = kk * 4 + koff;
    v2f aw0 = fragRK(hl, lm, kb, FS);        // weights rows 0..15
    v2f aw1 = fragRK(hl, 16 + lm, kb, FS);   // rows 16..23 (+zeros)
    v2f bf0 = fragRK(fl, lm, kb, FS);        // f^T cols 0..15
    v2f bf1 = fragRK(fl, 16 + lm, kb, FS);   // cols 16..23 (+zeros)
    s00 = wmma_f32(aw0, bf0, s00);
    s01 = wmma_f32(aw0, bf1, s01);
    s10 = wmma_f32(aw1, bf0, s10);
    s11 = wmma_f32(aw1, bf1, s11);
  }

  // S tiles -> LDS scratch (overlaid in hbuf: weights fully consumed above)
  float* sl = hl;          // 24 x SS floats (<= 648)
  float* pl = hl + 1024;   // 24 x WS floats (<= 648)
  #pragma unroll
  for (int r = 0; r < 8; ++r) {
    int m0 = half * 8 + r;
    int m1 = 16 + half * 8 + r;
    int c0 = lm;
    int c1 = 16 + lm;
    sl[m0 * SS + c0] = s00[r];
    if (c1 < N_KP) sl[m0 * SS + c1] = s01[r];
    if (m1 < N_KP) {
      sl[m1 * SS + c0] = s10[r];
      if (c1 < N_KP) sl[m1 * SS + c1] = s11[r];
    }
  }

  // ---- Stage 4: row softmax, one lane per row ----
  if (lane < N_KP) {
    float v[N_KP];
    float mx = -INFINITY;
    #pragma unroll
    for (int c = 0; c < N_KP; ++c) {
      v[c] = sl[lane * SS + c];
      mx = fmaxf(mx, v[c]);
    }
    float sum = 0.0f;
    #pragma unroll
    for (int c = 0; c < N_KP; ++c) {
      v[c] = __expf(v[c] - mx);
      sum += v[c];
    }
    float inv = 1.0f / sum;
    #pragma unroll
    for (int c = 0; c < N_KP; ++c) pl[lane * WS + c] = v[c] * inv;
  }

  // ---- Stage 5: out = P @ f, store directly from C layout (coalesced) ----
  {
    v2f a5[2][6];
    #pragma unroll
    for (int mt = 0; mt < 2; ++mt)
      #pragma unroll
      for (int kk = 0; kk < 6; ++kk)
        a5[mt][kk] = fragRK(pl, mt * 16 + lm, kk * 4 + koff, WS);

    float* go = O + (size_t)b * (N_KP * DDIM);
    for (int nt = 0; nt < 8; ++nt) {
      int col = nt * 16 + lm;
      #pragma unroll
      for (int mt = 0; mt < 2; ++mt) {
        v8f acc = {};
        #pragma unroll
        for (int kk = 0; kk < 6; ++kk) {
          int kb = kk * 4 + koff;
          v2f bf;
          bf.x = fl[kb * FS + col];
          bf.y = fl[(kb + 1) * FS + col];
          acc = wmma_f32(a5[mt][kk], bf, acc);
        }
        #pragma unroll
        for (int r = 0; r < 8; ++r) {
          int m = mt * 16 + half * 8 + r;
          if (m < N_KP) go[m * DDIM + col] = acc[r];
        }
      }
    }
  }
}

extern "C" void kernel_launch(void* const* d_in, const int* in_sizes, int n_in,
                              void* d_out, int out_size, void* d_ws, size_t ws_size,
                              hipStream_t stream) {
  (void)d_ws; (void)ws_size; (void)n_in; (void)out_size;
  const float* F  = (const float*)d_in[0];  // [B,24,128]
  const float* W1 = (const float*)d_in[1];  // [24,24]
  const float* B1 = (const float*)d_in[2];  // [24]
  const float* W2 = (const float*)d_in[3];  // [24,24]
  const float* B2 = (const float*)d_in[4];  // [24]
  float* O = (float*)d_out;

  const int nbatch = in_sizes[0] / (N_KP * DDIM);
  const int nblocks = (nbatch + 1) / 2;  // 2 waves per block, 1 batch item per wave
  attn24_kernel<<<nblocks, 64, 0, stream>>>(F, W1, B1, W2, B2, O, nbatch);
}